// HyperbolicAggregation_54039278518949
// MI455X (gfx1250) — compile-verified
//
#include <hip/hip_runtime.h>
#include <math.h>
#include <stdint.h>

typedef float v2f __attribute__((ext_vector_type(2)));
typedef float v8f __attribute__((ext_vector_type(8)));

#define N_ROWS 16384
#define DIM 32
#define HEPS 1e-7f
#define HMAX_NORM (1.0f - 1e-5f)
#define KC 256          // K-chunk rows per LDS buffer (256*32*4 = 32 KB, x2 buffers)
#define BLOCK 256       // 8 waves of 32
#define WAVES 8

__device__ __forceinline__ void wait_asynccnt0() {
#if __has_builtin(__builtin_amdgcn_s_wait_asynccnt)
    __builtin_amdgcn_s_wait_asynccnt(0);
#else
    asm volatile("s_wait_asynccnt 0x0" ::: "memory");
#endif
}

// ---------------- Kernel 1: x_tangent = logmap0(x) ----------------
__global__ __launch_bounds__(BLOCK)
void hyp_tangent_kernel(const float* __restrict__ x, float* __restrict__ t) {
    const int wave = threadIdx.x >> 5;
    const int lane = threadIdx.x & 31;
    const int row  = blockIdx.x * WAVES + wave;     // one wave per row, lane = column
    float v = x[(size_t)row * DIM + lane];
    float s = v * v;
    #pragma unroll
    for (int m = 16; m >= 1; m >>= 1) s += __shfl_xor(s, m, 32);
    float nr = sqrtf(s);
    float ns = fmaxf(nr, HEPS);                     // safe norm
    float nc = fminf(ns, HMAX_NORM);                // keep arctanh finite
    float at = 0.5f * logf((1.0f + nc) / (1.0f - nc));  // atanh(nc)
    t[(size_t)row * DIM + lane] = at * v / ns;
}

// ---- Kernel 2: WMMA spmm (async-LDS double-buffered B) + degree norm + expmap0 + proj ----
__global__ __launch_bounds__(BLOCK)
void hyp_agg_kernel(const float* __restrict__ adj,
                    const float* __restrict__ t,
                    float* __restrict__ out) {
    __shared__ float lds[2][KC * DIM];              // 2 x 32 KB B double-buffer

    const int tid  = threadIdx.x;
    const int wave = tid >> 5;
    const int lane = tid & 31;
    const int half = lane >> 4;                     // lane group: 0 or 1
    const int l15  = lane & 15;
    const int row0 = blockIdx.x * (WAVES * 16) + wave * 16;   // 16-row strip per wave
    const int rowA = row0 + l15;                    // adj row this lane streams
    const size_t arow = (size_t)rowA * N_ROWS;

    const unsigned lds_base = (unsigned)(uintptr_t)(&lds[0][0]);  // flat low-32 = LDS offset
    const unsigned long long tbase = (unsigned long long)(uintptr_t)t;

    // Issue the async memory->LDS copies for chunk k0 into buffer `buf`
    // (GLOBAL_LOAD_ASYNC_TO_LDS_B128: no VGPR data, tracked by ASYNCcnt)
    auto stage_async = [&](int k0, int buf) {
        #pragma unroll
        for (int i = 0; i < (KC * DIM / 4) / BLOCK; ++i) {      // 8 x b128 per thread
            unsigned off16 = (unsigned)(tid + i * BLOCK) * 16u;
            unsigned dst = lds_base + (unsigned)buf * (KC * DIM * 4u) + off16;
            unsigned long long src = tbase + (unsigned long long)k0 * (DIM * 4ull) + off16;
            asm volatile("global_load_async_to_lds_b128 %0, %1, off"
                         :: "v"(dst), "v"(src) : "memory");
        }
    };

    v8f c0 = {};                                    // output cols 0..15
    v8f c1 = {};                                    // output cols 16..31
    float rsum = 0.0f;                              // degree accumulator

    // Prologue: fill buffer 0
    stage_async(0, 0);
    wait_asynccnt0();
    __syncthreads();

    int buf = 0;
    for (int k0 = 0; k0 < N_ROWS; k0 += KC, buf ^= 1) {
        if (k0 + KC < N_ROWS) {
            stage_async(k0 + KC, buf ^ 1);          // prefill next buffer, overlapped
            __builtin_prefetch(&adj[arow + k0 + KC + half * 2], 0, 3); // prime adj stream
        }

        const float* cur = &lds[buf][0];
        #pragma unroll 4
        for (int kk = 0; kk < KC; kk += 4) {
            // A tile 16x4 f32 (ISA layout): lanes0-15 K={k,k+1}, lanes16-31 K={k+2,k+3}
            v2f a = *(const v2f*)(adj + arow + (size_t)(k0 + kk + half * 2));
            rsum += a.x + a.y;                      // row degree, free from A regs

            // B tiles 4x16: VGPR j holds K-rows 2j (lanes0-15) / 2j+1 (lanes16-31)
            v2f b0, b1;
            b0.x = cur[(kk + half) * DIM + l15];
            b0.y = cur[(kk + 2 + half) * DIM + l15];
            b1.x = cur[(kk + half) * DIM + 16 + l15];
            b1.y = cur[(kk + 2 + half) * DIM + 16 + l15];

            c0 = __builtin_amdgcn_wmma_f32_16x16x4_f32(false, a, false, b0,
                                                       (short)0, c0, false, false);
            c1 = __builtin_amdgcn_wmma_f32_16x16x4_f32(false, a, false, b1,
                                                       (short)0, c1, false, false);
        }

        wait_asynccnt0();                           // my async copies landed
        __syncthreads();                            // everyone's copies landed / reads done
    }

    // Fold the two K-halves: lane l now holds the full degree of row (l & 15)
    rsum += __shfl_xor(rsum, 16, 32);

    // Epilogue: per C/D VGPR i, row = i (lanes 0-15) or i+8 (lanes 16-31)
    #pragma unroll
    for (int i = 0; i < 8; ++i) {
        const int mrow = i + half * 8;
        float dlo = __shfl(rsum, i, 32);
        float dhi = __shfl(rsum, i + 8, 32);
        float deg = half ? dhi : dlo;

        float u0 = c0[i] / deg;                     // aggregation (degree-normalized)
        float u1 = c1[i] / deg;

        float s = u0 * u0 + u1 * u1;                // row-norm over 32 cols:
        #pragma unroll                              // 16-lane xor reduce stays in half
        for (int m = 8; m >= 1; m >>= 1) s += __shfl_xor(s, m, 32);
        float nr = sqrtf(s);
        float nu = fmaxf(nr, HEPS);                 // safe norm
        float th = tanhf(nu);
        float f  = th / nu;                         // expmap0 scale
        float ne = fmaxf(f * nr, HEPS);             // ||expmap0(u)||
        float sc = fminf(1.0f, HMAX_NORM / ne);     // proj scale
        float g  = f * sc;

        out[(size_t)(row0 + mrow) * DIM + l15]      = u0 * g;
        out[(size_t)(row0 + mrow) * DIM + 16 + l15] = u1 * g;
    }
}

// ---------------- Host launcher ----------------
extern "C" void kernel_launch(void* const* d_in, const int* in_sizes, int n_in,
                              void* d_out, int out_size, void* d_ws, size_t ws_size,
                              hipStream_t stream) {
    const float* x   = (const float*)d_in[0];   // [N, 32] f32
    const float* adj = (const float*)d_in[1];   // [N, N] f32
    float* out = (float*)d_out;                 // [N, 32] f32
    float* t   = (float*)d_ws;                  // [N, 32] f32 tangent scratch (2 MB)

    hyp_tangent_kernel<<<N_ROWS / WAVES, BLOCK, 0, stream>>>(x, t);
    hyp_agg_kernel<<<N_ROWS / (WAVES * 16), BLOCK, 0, stream>>>(adj, t, out);
}